// NashLoss3D_7301444403972
// MI455X (gfx1250) — compile-verified
//
#include <hip/hip_runtime.h>

// ---------------------------------------------------------------------------
// NashLoss3D for MI455X (gfx1250, wave32).
// pred/targ: float32 [nt=2000, ns=5000, ny=8], contiguous, ny innermost.
//
// Stage 1: streaming t-reduction, bandwidth-bound (640 MB @ 23.3 TB/s floor).
//   Each wave owns 32 consecutive columns as TWO 16-column WMMA tiles
//   (even columns = tile E, odd = tile O). Lane n loads float2 (b64) at
//   col_base+2n -> each half-wave request is a full 128 B line. Each 4-row
//   step feeds V_WMMA_F32_16X16X4_F32 with A = ones, so the matrix pipe
//   accumulates (cnt, sum_t, sum_t2, res) over t for all 32 columns.
// Stage 2: per-y combine of chunk partials + guards (deterministic LDS tree).
// Stage 3: scalar finalize.
// ---------------------------------------------------------------------------

typedef __attribute__((ext_vector_type(2))) float v2f;
typedef __attribute__((ext_vector_type(8))) float v8f;

#define NT_DIM        2000
#define NS_DIM        5000
#define NY_DIM        8
#define NCOL          (NS_DIM * NY_DIM)        // 40000 columns (s*8 + y)
#define NCHUNK        10
#define TCHUNK        (NT_DIM / NCHUNK)        // 200 (multiple of 4)
#define COLS_PER_WAVE 32                       // two 16-col WMMA tiles
#define WAVES_PER_BLK 2
#define COLS_PER_BLK  (COLS_PER_WAVE * WAVES_PER_BLK)   // 64
#define NBLK_X        (NCOL / COLS_PER_BLK)             // 625 (exact)
#define MIN_COUNT     10.0f

__device__ __forceinline__ void nash_elem(float g, float p,
                                          float& m, float& tc,
                                          float& t2, float& r)
{
    const bool o = (g == g);          // observed (not NaN)
    m  = o ? 1.0f : 0.0f;
    tc = o ? g : 0.0f;                // t_clean
    t2 = tc * tc;
    const float d = tc - p;
    r  = m * d * d;                   // masked residual^2
}

// Stage 1: per (chunk, column) partials {cnt, sum_t, sum_t2, res}.
__global__ __launch_bounds__(64) void nash_partial_kernel(
    const float* __restrict__ pred,
    const float* __restrict__ targ,
    float4* __restrict__ part)
{
    const int lane = threadIdx.x & 31;
    const int wave = threadIdx.x >> 5;
    const int n    = lane & 15;          // B-matrix N slot
    const int half = lane >> 4;          // K-slot group (rows 2h, 2h+1)
    const int colb = blockIdx.x * COLS_PER_BLK + wave * COLS_PER_WAVE;
    const int c2   = colb + 2 * n;       // even column owned by this lane
    const int t0   = blockIdx.y * TCHUNK;

    // Per-lane base: rows advance via immediate offsets inside the loop.
    const size_t base = (size_t)(t0 + 2 * half) * NCOL + c2;
    const float* pp = pred + base;
    const float* tp = targ + base;

    v2f ones; ones[0] = 1.0f; ones[1] = 1.0f;   // A = 16x4 all-ones
    v8f cE_cnt = {}, cE_st = {}, cE_st2 = {}, cE_res = {};
    v8f cO_cnt = {}, cO_st = {}, cO_st2 = {}, cO_res = {};

    #pragma unroll 2
    for (int tt = 0; tt < TCHUNK; tt += 4) {
        const size_t o0 = (size_t)tt * NCOL;
        // 128 B per half-wave per instruction (16 lanes x 8 B contiguous).
        const v2f p0 = __builtin_nontemporal_load((const v2f*)(pp + o0));
        const v2f p1 = __builtin_nontemporal_load((const v2f*)(pp + o0 + NCOL));
        const v2f g0 = __builtin_nontemporal_load((const v2f*)(tp + o0));
        const v2f g1 = __builtin_nontemporal_load((const v2f*)(tp + o0 + NCOL));

        float m0e, t0e, q0e, r0e, m1e, t1e, q1e, r1e;
        float m0o, t0o, q0o, r0o, m1o, t1o, q1o, r1o;
        nash_elem(g0[0], p0[0], m0e, t0e, q0e, r0e);   // row 2h,   even col
        nash_elem(g1[0], p1[0], m1e, t1e, q1e, r1e);   // row 2h+1, even col
        nash_elem(g0[1], p0[1], m0o, t0o, q0o, r0o);   // row 2h,   odd col
        nash_elem(g1[1], p1[1], m1o, t1o, q1o, r1o);   // row 2h+1, odd col

        v2f bmE;  bmE[0]  = m0e; bmE[1]  = m1e;
        v2f btE;  btE[0]  = t0e; btE[1]  = t1e;
        v2f bqE;  bqE[0]  = q0e; bqE[1]  = q1e;
        v2f brE;  brE[0]  = r0e; brE[1]  = r1e;
        v2f bmO;  bmO[0]  = m0o; bmO[1]  = m1o;
        v2f btO;  btO[0]  = t0o; btO[1]  = t1o;
        v2f bqO;  bqO[0]  = q0o; bqO[1]  = q1o;
        v2f brO;  brO[0]  = r0o; brO[1]  = r1o;

        // D = ones(16x4) x B(4x16) + C : every row of D = column sums.
        cE_cnt = __builtin_amdgcn_wmma_f32_16x16x4_f32(
            false, ones, false, bmE, (short)0, cE_cnt, false, false);
        cE_st  = __builtin_amdgcn_wmma_f32_16x16x4_f32(
            false, ones, false, btE, (short)0, cE_st,  false, false);
        cE_st2 = __builtin_amdgcn_wmma_f32_16x16x4_f32(
            false, ones, false, bqE, (short)0, cE_st2, false, false);
        cE_res = __builtin_amdgcn_wmma_f32_16x16x4_f32(
            false, ones, false, brE, (short)0, cE_res, false, false);
        cO_cnt = __builtin_amdgcn_wmma_f32_16x16x4_f32(
            false, ones, false, bmO, (short)0, cO_cnt, false, false);
        cO_st  = __builtin_amdgcn_wmma_f32_16x16x4_f32(
            false, ones, false, btO, (short)0, cO_st,  false, false);
        cO_st2 = __builtin_amdgcn_wmma_f32_16x16x4_f32(
            false, ones, false, bqO, (short)0, cO_st2, false, false);
        cO_res = __builtin_amdgcn_wmma_f32_16x16x4_f32(
            false, ones, false, brO, (short)0, cO_res, false, false);
    }

    // C/D layout: VGPR0, lanes 0..15 = row M=0, N=lane; all rows equal.
    if (half == 0) {
        const size_t o = (size_t)blockIdx.y * NCOL + c2;
        float4 ve, vo;
        ve.x = cE_cnt[0]; ve.y = cE_st[0]; ve.z = cE_st2[0]; ve.w = cE_res[0];
        vo.x = cO_cnt[0]; vo.y = cO_st[0]; vo.z = cO_st2[0]; vo.w = cO_res[0];
        part[o]     = ve;   // even column
        part[o + 1] = vo;   // odd column
    }
}

// Stage 2: one block per y. Combine chunk partials, apply guards, reduce over s.
__global__ __launch_bounds__(256) void nash_finalize_y(
    const float4* __restrict__ part,
    float* __restrict__ pery)
{
    __shared__ float s_term[256];
    __shared__ float s_cnt[256];

    const int y = blockIdx.x;   // 0..7
    float tsum = 0.0f;
    float nsum = 0.0f;

    for (int s = threadIdx.x; s < NS_DIM; s += 256) {
        const int col = s * NY_DIM + y;
        float cnt = 0.0f, st = 0.0f, st2 = 0.0f, res = 0.0f;
        #pragma unroll
        for (int k = 0; k < NCHUNK; ++k) {
            const float4 v = part[(size_t)k * NCOL + col];
            cnt += v.x; st += v.y; st2 += v.z; res += v.w;
        }
        const float cs   = fmaxf(cnt, 1.0f);
        const float mean = st / cs;
        // sst = sum m*(t-mean)^2 = st2 - 2*mean*st + cnt*mean^2
        const float sst  = st2 - 2.0f * mean * st + cnt * mean * mean;
        const bool valid = (cnt > MIN_COUNT) && (sst != 0.0f);
        const float ss   = valid ? sst : 1.0f;
        const float den  = sqrtf(ss) + 0.1f;
        const float term = valid ? res / (den * den) : 0.0f;
        tsum += term;
        nsum += valid ? 1.0f : 0.0f;
    }

    s_term[threadIdx.x] = tsum;
    s_cnt[threadIdx.x]  = nsum;
    __syncthreads();
    for (int o = 128; o > 0; o >>= 1) {
        if ((int)threadIdx.x < o) {
            s_term[threadIdx.x] += s_term[threadIdx.x + o];
            s_cnt[threadIdx.x]  += s_cnt[threadIdx.x + o];
        }
        __syncthreads();
    }
    if (threadIdx.x == 0) {
        const float nv = s_cnt[0];
        pery[y] = (nv > 0.0f) ? (s_term[0] / fmaxf(nv, 1.0f)) : 0.0f;
    }
}

// Stage 3: scalar.
__global__ void nash_final(const float* __restrict__ pery,
                           float* __restrict__ out)
{
    if (threadIdx.x == 0 && blockIdx.x == 0) {
        float s = 0.0f;
        #pragma unroll
        for (int y = 0; y < NY_DIM; ++y) s += pery[y];
        out[0] = s * (1.0f / (float)NY_DIM);
    }
}

extern "C" void kernel_launch(void* const* d_in, const int* in_sizes, int n_in,
                              void* d_out, int out_size, void* d_ws, size_t ws_size,
                              hipStream_t stream) {
    (void)in_sizes; (void)n_in; (void)out_size; (void)ws_size;
    const float* pred = (const float*)d_in[0];
    const float* targ = (const float*)d_in[1];

    float4* part = (float4*)d_ws;                       // NCHUNK*NCOL float4 = 6.4 MB
    float*  pery = (float*)((char*)d_ws +
                            sizeof(float4) * (size_t)NCHUNK * NCOL);  // 8 floats

    dim3 grid1(NBLK_X, NCHUNK);
    nash_partial_kernel<<<grid1, 64, 0, stream>>>(pred, targ, part);
    nash_finalize_y<<<NY_DIM, 256, 0, stream>>>(part, pery);
    nash_final<<<1, 32, 0, stream>>>(pery, (float*)d_out);
}